// EnDecoder_26603027432101
// MI455X (gfx1250) — compile-verified
//
#include <hip/hip_runtime.h>

typedef __attribute__((ext_vector_type(2))) float v2f;
typedef __attribute__((ext_vector_type(8))) float v8f;

#define NJ   22
#define NC   66      // NJ*3 skeleton components
#define KD   10      // betas dim
#define SROW 67      // LDS row stride (odd -> no bank conflicts)
#define SPB  128     // samples per block == threads per block (4 waves)

__global__ __launch_bounds__(SPB)
void fk_fused_kernel(const float* __restrict__ body_pose,     // (ns,63)
                     const float* __restrict__ betas,         // (ns,10)
                     const float* __restrict__ global_orient, // (ns,3)
                     const float* __restrict__ transl,        // (ns,3)
                     const float* __restrict__ J_template,    // (22,3) = 66
                     const float* __restrict__ J_shapedirs,   // (22,3,10) = (n,k)
                     float* __restrict__ out,                 // (ns,22,3)
                     int ns)
{
    __shared__ float smem[SPB * SROW];

    const int tid  = threadIdx.x;
    const int lane = tid & 31;
    const int wave = tid >> 5;
    const int m    = lane & 15;   // matrix row/col within half-wave
    const int kh   = lane >> 4;   // 0: lanes 0-15, 1: lanes 16-31
    const int blockBase = blockIdx.x * SPB;

    // ---------------- Phase 1: skeleton GEMM via V_WMMA_F32_16X16X4_F32 ----
    // skeleton(16 samples x 80 comps) = betas(16x12) @ shapedirs^T(12x80) + template
    // B fragments (shapedirs^T) + template: identical for both 16-sample halves.
    v2f  bf[5][3];
    float tv[5];
#pragma unroll
    for (int t = 0; t < 5; ++t) {
        const int n = t * 16 + m;                  // skeleton component column
        tv[t] = (n < NC) ? J_template[n] : 0.0f;
#pragma unroll
        for (int ks = 0; ks < 3; ++ks) {
            const int k0 = ks * 4 + 2 * kh;        // 32-bit B layout: K split by lane half
            const float bx = (n < NC && k0     < KD) ? J_shapedirs[n * KD + k0    ] : 0.0f;
            const float by = (n < NC && k0 + 1 < KD) ? J_shapedirs[n * KD + k0 + 1] : 0.0f;
            bf[t][ks].x = bx; bf[t][ks].y = by;
        }
    }

#pragma unroll
    for (int h = 0; h < 2; ++h) {
        const int sLocal = wave * 32 + h * 16;     // first of this half's 16 samples
        // A fragments: 16 beta rows, 32-bit A layout (VGPR0: K0|K2, VGPR1: K1|K3)
        v2f af[3];
#pragma unroll
        for (int ks = 0; ks < 3; ++ks) {
            const int k0 = ks * 4 + 2 * kh;
            int srow = blockBase + sLocal + m;
            srow = (srow < ns) ? srow : (ns - 1);  // clamp (last block may extend past ns)
            const float axv = (k0     < KD) ? betas[(size_t)srow * KD + k0    ] : 0.0f;
            const float ayv = (k0 + 1 < KD) ? betas[(size_t)srow * KD + k0 + 1] : 0.0f;
            af[ks].x = axv; af[ks].y = ayv;
        }
#pragma unroll
        for (int t = 0; t < 5; ++t) {
            v8f acc = {};
#pragma unroll
            for (int ks = 0; ks < 3; ++ks) {
                acc = __builtin_amdgcn_wmma_f32_16x16x4_f32(
                    /*neg_a=*/false, af[ks],
                    /*neg_b=*/false, bf[t][ks],
                    /*c_mod=*/(short)0, acc,
                    /*reuse_a=*/false, /*reuse_b=*/false);
            }
            const int n = t * 16 + m;
#pragma unroll
            for (int r = 0; r < 8; ++r) {
                const int M = r + 8 * kh;          // C layout: lanes16-31 hold rows 8-15
                if (n < NC)
                    smem[(sLocal + M) * SROW + n] = acc[r] + tv[t];
            }
        }
    }
    __syncthreads();

    // ---------------- Phase 2: Rodrigues + FK chain, one sample per lane ----
    const int s = blockBase + tid;
    if (s >= ns) return;

    const float* __restrict__ sk = &smem[tid * SROW];          // skeleton(66)
    const float* __restrict__ bp = body_pose     + (size_t)s * 63;
    const float* __restrict__ go = global_orient + (size_t)s * 3;
    const float* __restrict__ tr = transl        + (size_t)s * 3;
    float*       __restrict__ op = out           + (size_t)s * NC;

    constexpr int par[NJ] = {-1,0,0,0,1,2,3,4,5,6,7,8,9,9,9,12,13,14,16,17,18,19};

    float Rg[NJ][9];
    float Px[NJ], Py[NJ], Pz[NJ];

#pragma unroll
    for (int j = 0; j < NJ; ++j) {
        // axis-angle -> rotation (Rodrigues), matching reference's 1e-12 guard
        float ax, ay, az;
        if (j == 0) { ax = go[0]; ay = go[1]; az = go[2]; }
        else        { ax = bp[3*(j-1)]; ay = bp[3*(j-1)+1]; az = bp[3*(j-1)+2]; }
        const float sq   = ax*ax + ay*ay + az*az + 1e-12f;
        const float rinv = rsqrtf(sq);
        const float ang  = sq * rinv;              // == sqrt(sq)
        const float ux = ax*rinv, uy = ay*rinv, uz = az*rinv;
        const float cA = __cosf(ang), sA = __sinf(ang), tA = 1.0f - cA;
        const float r00 = tA*ux*ux + cA,    r01 = tA*ux*uy - sA*uz, r02 = tA*ux*uz + sA*uy;
        const float r10 = tA*ux*uy + sA*uz, r11 = tA*uy*uy + cA,    r12 = tA*uy*uz - sA*ux;
        const float r20 = tA*ux*uz - sA*uy, r21 = tA*uy*uz + sA*ux, r22 = tA*uz*uz + cA;

        if (j == 0) {
            Rg[0][0]=r00; Rg[0][1]=r01; Rg[0][2]=r02;
            Rg[0][3]=r10; Rg[0][4]=r11; Rg[0][5]=r12;
            Rg[0][6]=r20; Rg[0][7]=r21; Rg[0][8]=r22;
            Px[0] = sk[0] + tr[0];
            Py[0] = sk[1] + tr[1];
            Pz[0] = sk[2] + tr[2];
        } else {
            const int p = par[j];
            const float tx = sk[3*j]   - sk[3*p];
            const float ty = sk[3*j+1] - sk[3*p+1];
            const float tz = sk[3*j+2] - sk[3*p+2];
            const float* __restrict__ Pm = Rg[p];
            Px[j] = Pm[0]*tx + Pm[1]*ty + Pm[2]*tz + Px[p];
            Py[j] = Pm[3]*tx + Pm[4]*ty + Pm[5]*tz + Py[p];
            Pz[j] = Pm[6]*tx + Pm[7]*ty + Pm[8]*tz + Pz[p];
            Rg[j][0] = Pm[0]*r00 + Pm[1]*r10 + Pm[2]*r20;
            Rg[j][1] = Pm[0]*r01 + Pm[1]*r11 + Pm[2]*r21;
            Rg[j][2] = Pm[0]*r02 + Pm[1]*r12 + Pm[2]*r22;
            Rg[j][3] = Pm[3]*r00 + Pm[4]*r10 + Pm[5]*r20;
            Rg[j][4] = Pm[3]*r01 + Pm[4]*r11 + Pm[5]*r21;
            Rg[j][5] = Pm[3]*r02 + Pm[4]*r12 + Pm[5]*r22;
            Rg[j][6] = Pm[6]*r00 + Pm[7]*r10 + Pm[8]*r20;
            Rg[j][7] = Pm[6]*r01 + Pm[7]*r11 + Pm[8]*r21;
            Rg[j][8] = Pm[6]*r02 + Pm[7]*r12 + Pm[8]*r22;
        }
        op[3*j]   = Px[j];
        op[3*j+1] = Py[j];
        op[3*j+2] = Pz[j];
    }
}

extern "C" void kernel_launch(void* const* d_in, const int* in_sizes, int n_in,
                              void* d_out, int out_size, void* d_ws, size_t ws_size,
                              hipStream_t stream) {
    const float* body_pose     = (const float*)d_in[0];
    const float* betas         = (const float*)d_in[1];
    const float* global_orient = (const float*)d_in[2];
    const float* transl        = (const float*)d_in[3];
    const float* J_template    = (const float*)d_in[4];
    const float* J_shapedirs   = (const float*)d_in[5];

    const int ns   = in_sizes[0] / 63;      // B*L samples
    const int grid = (ns + SPB - 1) / SPB;

    fk_fused_kernel<<<grid, SPB, 0, stream>>>(
        body_pose, betas, global_orient, transl, J_template, J_shapedirs,
        (float*)d_out, ns);
}